// SoftDTW_3135326126531
// MI455X (gfx1250) — compile-verified
//
#include <hip/hip_runtime.h>
#include <hip/hip_bf16.h>
#include <math.h>

// ---------------------------------------------------------------------------
// Soft-DTW, B=128, N=M=256, d=8, gamma=1, BIG=1e10
//   Kernel 1: D[b,n,m] = |x_n|^2 + |y_m|^2 - 2 x_n.y_m  via V_WMMA_F32_16X16X4_F32
//   Kernel 2: anti-diagonal wavefront DP, one workgroup per batch (LDS diagonals)
// Workspace: B*N*M*4 = 32 MB (fits in d_ws; lives in 192MB L2 between kernels)
// ---------------------------------------------------------------------------

#define BATCH 128
#define NN    256
#define MM    256
#define DD    8
#define BIGV  1.0e10f

typedef __attribute__((ext_vector_type(2))) float v2f;
typedef __attribute__((ext_vector_type(8))) float v8f;

// ---------- Kernel 1: distance matrix via WMMA (one wave per 16x16 tile) ----
__global__ __launch_bounds__(32)
void softdtw_dist_wmma(const float* __restrict__ x,
                       const float* __restrict__ y,
                       float* __restrict__ Dm) {
    const int b  = blockIdx.x;       // batch
    const int ti = blockIdx.y;       // x-row tile (n)
    const int tj = blockIdx.z;       // y-row tile (m)
    const int lane = threadIdx.x;    // 0..31 (wave32)
    const int half = lane >> 4;      // 0 or 1
    const int idx  = lane & 15;      // 0..15

    // A fragment (x rows, 16x4) and B fragment (y rows^T, 4x16):
    // element (vgpr v, lane L) = [row = L%16, k = 2*(L/16) + v]
    const float* xr = x + ((size_t)b * NN + (size_t)ti * 16 + idx) * DD;
    const float* yr = y + ((size_t)b * MM + (size_t)tj * 16 + idx) * DD;

    v2f a0, a1, b0, b1;
    a0[0] = xr[2 * half + 0];     a0[1] = xr[2 * half + 1];      // k = 0..3
    a1[0] = xr[4 + 2 * half + 0]; a1[1] = xr[4 + 2 * half + 1];  // k = 4..7
    b0[0] = yr[2 * half + 0];     b0[1] = yr[2 * half + 1];
    b1[0] = yr[4 + 2 * half + 0]; b1[1] = yr[4 + 2 * half + 1];

    v8f acc = {};
    // acc = x . y^T  over K=8 as two K=4 WMMAs (EXEC all-ones: no divergence yet)
    acc = __builtin_amdgcn_wmma_f32_16x16x4_f32(false, a0, false, b0,
                                                (short)0, acc, false, false);
    acc = __builtin_amdgcn_wmma_f32_16x16x4_f32(false, a1, false, b1,
                                                (short)0, acc, false, false);

    // Squared norms of the 16 x-rows / 16 y-rows of this tile, via LDS.
    __shared__ float sx2[16];
    __shared__ float sy2[16];
    {
        const float* r = (lane < 16) ? xr : yr;  // lane<16 did x-row `idx`, else y-row `idx`
        float s = 0.f;
#pragma unroll
        for (int d = 0; d < DD; ++d) s += r[d] * r[d];
        if (lane < 16) sx2[idx] = s; else sy2[idx] = s;
    }
    __syncthreads();

    // C/D layout: lane L, vgpr v -> (M = v + 8*(L/16), N = L%16)
    const float y2v = sy2[idx];
    float* outRow = Dm + (((size_t)b * NN + (size_t)ti * 16) * MM) + (size_t)tj * 16 + idx;
#pragma unroll
    for (int v = 0; v < 8; ++v) {
        const int m = v + 8 * half;                 // local x-row
        const float dij = fmaf(-2.0f, acc[v], sx2[m] + y2v);
        outRow[(size_t)m * MM] = dij;
    }
}

// ---------- Kernel 2: wavefront DP, one 256-thread block per batch ----------
__global__ __launch_bounds__(256)
void softdtw_dp(const float* __restrict__ Dm, float* __restrict__ out) {
    const int b = blockIdx.x;
    const float* Db = Dm + (size_t)b * NN * MM;

    // Three rotating diagonals R[k-2], R[k-1], R[k]; index i = 0..N (pad width)
    __shared__ float R[3][NN + 8];

    const int t = threadIdx.x;   // 0..255
    const int i = t + 1;         // DP row 1..256

    // diag k=0: R[0][0]=0, rest BIG ; diag k=1: all BIG
    if (t == 0) { R[0][0] = 0.0f; R[1][0] = BIGV; }
    R[0][i] = BIGV;
    R[1][i] = BIGV;
    __syncthreads();

    int p2 = 0, p1 = 1, pc = 2;
    const float* Drow = Db + (size_t)(i - 1) * MM;

#pragma unroll 1
    for (int k = 2; k <= NN + MM; ++k) {
        const int j = k - i;
        float v = BIGV;
        if ((j >= 1) && (j <= MM)) {
            const float rd = R[p2][i - 1];   // R[i-1, j-1]
            const float ru = R[p1][i - 1];   // R[i-1, j  ]
            const float rl = R[p1][i];       // R[i  , j-1]
            const float mn = fminf(rd, fminf(ru, rl));
            // softmin = mn - gamma*log(sum exp((mn - r)/gamma)), gamma = 1
            const float s = __expf(mn - rd) + __expf(mn - ru) + __expf(mn - rl);
            v = Drow[j - 1] + (mn - __logf(s));
            if (j < MM) __builtin_prefetch(&Drow[j], 0, 1);  // next diagonal's cost
        }
        R[pc][i] = v;
        if (t == 0) R[pc][0] = BIGV;          // border cell (0, k)
        __syncthreads();
        const int tmp = p2; p2 = p1; p1 = pc; pc = tmp;
    }

    // last diagonal (k = N+M) now lives in p1; result is R[N, M] at i = N
    if (i == NN) out[b] = R[p1][NN];
}

// ---------------------------------------------------------------------------
extern "C" void kernel_launch(void* const* d_in, const int* in_sizes, int n_in,
                              void* d_out, int out_size, void* d_ws, size_t ws_size,
                              hipStream_t stream) {
    const float* x = (const float*)d_in[0];   // [128, 256, 8] f32
    const float* y = (const float*)d_in[1];   // [128, 256, 8] f32
    float* out = (float*)d_out;               // [128] f32
    float* Dws = (float*)d_ws;                // 128*256*256*4 = 32 MB scratch

    dim3 g1(BATCH, NN / 16, MM / 16);
    softdtw_dist_wmma<<<g1, 32, 0, stream>>>(x, y, Dws);

    softdtw_dp<<<BATCH, NN, 0, stream>>>(Dws, out);
}